// ELLGAT_18537078849856
// MI455X (gfx1250) — compile-verified
//
#include <hip/hip_runtime.h>
#include <math.h>

#define DEG    32
#define HEADS  4
#define OUT_F  16
#define IN_F   64
#define HF     (HEADS * OUT_F)   // 64 combined head*feature rows

typedef float v2f __attribute__((ext_vector_type(2)));
typedef float v8f __attribute__((ext_vector_type(8)));

// ---------------------------------------------------------------------------
// Kernel 1: fused projection GEMM via V_WMMA_F32_16X16X4_F32.
//   Out[r, n] = sum_i W[r, i] * Q[i, n]      (r = h*16+o, W is qw or kw)
// One wave computes one 16(M) x 16(N) tile, accumulating K=64 in steps of 4.
// Results stored node-major: Qo[n*64 + r], Ko[n*64 + r] so that phase 2 reads
// a neighbor's full feature row as one contiguous 256B block.
// ---------------------------------------------------------------------------
__global__ __launch_bounds__(32)
void ellgat_proj_wmma(const float* __restrict__ qw,
                      const float* __restrict__ kw,
                      const float* __restrict__ Q,
                      float* __restrict__ Qo,
                      float* __restrict__ Ko,
                      int N)
{
    const int lane  = threadIdx.x;          // 0..31, wave32
    const int ctile = blockIdx.x;           // column (node) tile
    const int mtile = blockIdx.y;           // 0..7: 0-3 -> Qo rows, 4-7 -> Ko rows

    const float* W   = (mtile < 4) ? (qw + (size_t)mtile * 16 * IN_F)
                                   : (kw + (size_t)(mtile - 4) * 16 * IN_F);
    float*       Out = (mtile < 4) ? Qo : Ko;
    const int rbase  = (mtile & 3) * 16;    // row offset inside 64-row output

    const int n0    = ctile * 16;
    const int l15   = lane & 15;
    const int khalf = (lane >> 4) << 1;     // 0 for lanes 0-15, 2 for lanes 16-31
    const int ncol  = min(n0 + l15, N - 1); // clamped load column

    v8f c = {0.f, 0.f, 0.f, 0.f, 0.f, 0.f, 0.f, 0.f};

    #pragma unroll
    for (int k0 = 0; k0 < IN_F; k0 += 4) {
        // A: 16x4 fp32 tile. Lane holds row M=l15; VGPR0/1 hold K = khalf, khalf+1.
        v2f a;
        a.x = W[l15 * IN_F + k0 + khalf];
        a.y = W[l15 * IN_F + k0 + khalf + 1];
        // B: 4x16 fp32 tile. Lanes 0-15 carry K rows 0/1, lanes 16-31 K rows 2/3.
        v2f b;
        b.x = Q[(size_t)(k0 + khalf)     * N + ncol];
        b.y = Q[(size_t)(k0 + khalf + 1) * N + ncol];
        // D = A*B + C  (8 args: neg_a, A, neg_b, B, c_mod, C, reuse_a, reuse_b)
        c = __builtin_amdgcn_wmma_f32_16x16x4_f32(false, a, false, b,
                                                  (short)0, c, false, false);
    }

    // C/D layout: VGPR v holds M = v (lanes 0-15) or v+8 (lanes 16-31), N = lane%16.
    const int n = n0 + l15;
    if (n < N) {
        const int mb = rbase + (lane >> 4) * 8;
        #pragma unroll
        for (int v = 0; v < 8; ++v)
            Out[(size_t)n * HF + mb + v] = c[v];
    }
}

// ---------------------------------------------------------------------------
// Kernel 2: per-node attention. One wave per node, lane = neighbor (DEG == 32).
//   logit[h,k] = sum_o a[h,o] * leaky_relu(Qo[h,o,n] + Ko[h,o,adj[n,k]])
//   softmax over k with -inf masking of padded (-1) neighbors
//   out[h,o,n] = sum_k attn[h,k] * Ko[h,o,adj[n,k]]
// ---------------------------------------------------------------------------
__global__ __launch_bounds__(256)
void ellgat_attn(const int*   __restrict__ adj,
                 const float* __restrict__ Qo,
                 const float* __restrict__ Ko,
                 const float* __restrict__ aw,
                 float*       __restrict__ out,
                 int N)
{
    const int lane = threadIdx.x & 31;
    const int wid  = threadIdx.x >> 5;
    const int n    = blockIdx.x * (blockDim.x >> 5) + wid;
    if (n >= N) return;                       // uniform per wave -> EXEC stays full

    const int  j     = adj[(size_t)n * DEG + lane];
    const bool valid = (j >= 0);
    const int  jj    = valid ? j : 0;         // clamped; weight forced to 0 below

    const float* kr = Ko + (size_t)jj * HF;   // neighbor feature row (256B)
    const float* qr = Qo + (size_t)n  * HF;   // own query row (wave-uniform)

    // ---- per-head logits -------------------------------------------------
    float logit[HEADS];
    #pragma unroll
    for (int h = 0; h < HEADS; ++h) {
        const float4* k4 = (const float4*)(kr + h * OUT_F);
        const float4* q4 = (const float4*)(qr + h * OUT_F);
        float acc = 0.f;
        #pragma unroll
        for (int t = 0; t < 4; ++t) {
            float4 kv = k4[t];
            float4 qv = q4[t];
            float sx = qv.x + kv.x, sy = qv.y + kv.y;
            float sz = qv.z + kv.z, sw = qv.w + kv.w;
            // leaky_relu(x) == max(x, 0.01*x) for slope in (0,1)
            sx = fmaxf(sx, 0.01f * sx);
            sy = fmaxf(sy, 0.01f * sy);
            sz = fmaxf(sz, 0.01f * sz);
            sw = fmaxf(sw, 0.01f * sw);
            acc += aw[h * OUT_F + t * 4 + 0] * sx
                 + aw[h * OUT_F + t * 4 + 1] * sy
                 + aw[h * OUT_F + t * 4 + 2] * sz
                 + aw[h * OUT_F + t * 4 + 3] * sw;
        }
        logit[h] = acc;
    }

    // ---- masked softmax across the wave (neighbors) ----------------------
    float wgt[HEADS];
    #pragma unroll
    for (int h = 0; h < HEADS; ++h) {
        float x = valid ? logit[h] : -__builtin_inff();
        float m = x;
        #pragma unroll
        for (int s = 16; s >= 1; s >>= 1)
            m = fmaxf(m, __shfl_xor(m, s, 32));
        float e = valid ? __expf(x - m) : 0.f;   // all-padded row -> e == 0
        float ssum = e;
        #pragma unroll
        for (int s = 16; s >= 1; s >>= 1)
            ssum += __shfl_xor(ssum, s, 32);
        wgt[h] = (ssum > 0.f) ? (e / ssum) : 0.f;
    }

    // ---- weighted aggregation, lanes span 2 heads x 16 features ----------
    // pair p covers output rows ho = p*32 + lane (contiguous 128B Ko slice).
    #pragma unroll
    for (int p = 0; p < 2; ++p) {
        const int base = p * 32;
        const float wlo = wgt[p * 2 + 0];
        const float whi = wgt[p * 2 + 1];
        float acc = 0.f;
        #pragma unroll 4
        for (int k = 0; k < DEG; ++k) {
            int   jk  = __shfl(jj,  k, 32);
            float wk0 = __shfl(wlo, k, 32);
            float wk1 = __shfl(whi, k, 32);
            float wk  = (lane < 16) ? wk0 : wk1;   // padded neighbors: wk == 0
            acc += wk * Ko[(size_t)jk * HF + base + lane];
        }
        // out is [H, O, N] flat: index ((h*16+o)*N + n) == (base+lane)*N + n
        out[(size_t)(base + lane) * N + n] = acc;
    }
}

extern "C" void kernel_launch(void* const* d_in, const int* in_sizes, int n_in,
                              void* d_out, int out_size, void* d_ws, size_t ws_size,
                              hipStream_t stream) {
    const int*   adj = (const int*)  d_in[0];   // [N, 32]
    const float* Q   = (const float*)d_in[1];   // [64, N]
    const float* qw  = (const float*)d_in[2];   // [4, 16, 64]
    const float* kw  = (const float*)d_in[3];   // [4, 16, 64]
    const float* aw  = (const float*)d_in[4];   // [4, 16]
    float*       out = (float*)d_out;           // [4, 16, N]

    const int N = in_sizes[1] / IN_F;

    float* Qo = (float*)d_ws;                   // [N, 64] node-major
    float* Ko = Qo + (size_t)N * HF;            // [N, 64] node-major

    dim3 pgrid((N + 15) / 16, 8);
    ellgat_proj_wmma<<<pgrid, 32, 0, stream>>>(qw, kw, Q, Qo, Ko, N);

    const int wavesPerBlock = 8;                // 256 threads
    dim3 agrid((N + wavesPerBlock - 1) / wavesPerBlock);
    ellgat_attn<<<agrid, wavesPerBlock * 32, 0, stream>>>(adj, Qo, Ko, aw, out, N);
}